// PoolingRNNGlobal_41291815583749
// MI455X (gfx1250) — compile-verified
//
#include <hip/hip_runtime.h>
#include <hip/hip_bf16.h>

// ---------------------------------------------------------------------------
// PoolingRNN (bidirectional tanh RNN + word-span pooling) for MI455X / gfx1250
//
// Phase 0: init_kernel  — zero d_out + h double buffers + barrier counters.
// Phase 1: xproj_kernel — XP[b][t][0:512]   = x@W_ih_f^T + b_ih_f + b_hh_f
//                         XP[b][t][512:1024]= x@W_ih_b^T + b_ih_b + b_hh_b
//            V_WMMA_F32_16X16X4_F32, A-tile (64x1024 f32) staged in LDS,
//            32x64 per-wave tile => ~4x less L2 traffic than naive tiling.
// Phase 2: scan_kernel  — 8 persistent WGs (4 per direction). Each WG holds a
//            128x512 f32 W_hh slice in LDS (256KB, DMA'd via TDM
//            tensor_load_to_lds when available), per-step WMMA + LDS reduce +
//            tanh + masked h update; pooled outputs written directly; one
//            4-participant device barrier per step (double-buffered h).
// ---------------------------------------------------------------------------

#define B_  8
#define T_  2048
#define I_  1024
#define H_  512
#define NW_ 1024
#define H2_ 1024

typedef float v2f __attribute__((ext_vector_type(2)));
typedef float v8f __attribute__((ext_vector_type(8)));

// ws layout (float units)
#define XP_FLOATS  (B_ * T_ * H2_)                 // 16,777,216 (64 MB)
#define HB_FLOATS  (2 * 2 * B_ * H_)               // [dir][buf][8][512]
#define CTR_OFF    (XP_FLOATS + HB_FLOATS)

__device__ __forceinline__ v8f wmma_f32(v2f a, v2f b, v8f c) {
  return __builtin_amdgcn_wmma_f32_16x16x4_f32(false, a, false, b,
                                               (short)0, c, false, false);
}

__device__ __forceinline__ float fast_tanh(float x) {
  float e = __expf(2.0f * x);          // v_exp_f32; saturates correctly
  return 1.0f - 2.0f / (e + 1.0f);
}

// ---------------------------------------------------------------------------
// TDM: DMA a 2D f32 tile (rows x cols, row stride == cols) global -> LDS.
// D# per CDNA5 ISA. Guarded: falls back to float4 copies.
// ---------------------------------------------------------------------------
#if __has_builtin(__builtin_amdgcn_tensor_load_to_lds)
#define HAVE_TDM 1
typedef unsigned int v4u __attribute__((ext_vector_type(4)));
typedef int v8i __attribute__((ext_vector_type(8)));
typedef int v4i __attribute__((ext_vector_type(4)));

__device__ __forceinline__ void tdm_load_2d_f32(const float* gsrc, float* lds_dst,
                                                unsigned rows, unsigned cols) {
  unsigned long long ga = (unsigned long long)(uintptr_t)gsrc;
  unsigned la = (unsigned)(uintptr_t)lds_dst;     // LDS aperture: addr[31:0]
  v4u g0;
  g0[0] = 1u;                                     // count=1 (valid user D#)
  g0[1] = la;                                     // lds_addr (bytes)
  g0[2] = (unsigned)ga;                           // global_addr[31:0]
  g0[3] = (unsigned)((ga >> 32) & 0x1FFFFFFu) | (2u << 30);  // addr[56:32]|type=2
  v8i g1;
  g1[0] = (int)(2u << 16);                        // data_size=2 (4 bytes)
  g1[1] = (int)((cols & 0xFFFFu) << 16);          // tensor_dim0[15:0]
  g1[2] = (int)(((cols >> 16) & 0xFFFFu) | ((rows & 0xFFFFu) << 16)); // d0 hi|d1 lo
  g1[3] = (int)(((rows >> 16) & 0xFFFFu) | ((cols & 0xFFFFu) << 16)); // d1 hi|tile_dim0
  g1[4] = (int)(rows & 0xFFFFu);                  // tile_dim1 | tile_dim2=0
  g1[5] = (int)cols;                              // tensor_dim0_stride[31:0]
  g1[6] = 0;                                      // stride hi | dim1_stride lo
  g1[7] = 0;
  v4i z4 = {0, 0, 0, 0};
#if defined(__clang_major__) && __clang_major__ >= 23
  v8i z8 = {0, 0, 0, 0, 0, 0, 0, 0};
  __builtin_amdgcn_tensor_load_to_lds(g0, g1, z4, z4, z8, 0);
#else
  __builtin_amdgcn_tensor_load_to_lds(g0, g1, z4, z4, 0);
#endif
}
#endif

// ---------------------------------------------------------------------------
// Phase 0: zero d_out + ws control region
// ---------------------------------------------------------------------------
__global__ void init_kernel(float4* __restrict__ out4, float* __restrict__ ws_misc) {
  unsigned idx = blockIdx.x * 256u + threadIdx.x;   // 8192*256 = 2,097,152
  out4[idx] = make_float4(0.f, 0.f, 0.f, 0.f);
  if (idx < HB_FLOATS + 64u) ws_misc[idx] = 0.f;
}

// ---------------------------------------------------------------------------
// Phase 1: input projection GEMM. grid = (256 M-blocks, 4 N-blocks), 256 thr.
// Block tile 64x256, A (64x1024 f32) staged in LDS, wave tile 32x64.
// ---------------------------------------------------------------------------
__global__ void xproj_kernel(const float* __restrict__ x,
                             const float* __restrict__ wf,
                             const float* __restrict__ wb,
                             const float* __restrict__ bihf,
                             const float* __restrict__ bhhf,
                             const float* __restrict__ bihb,
                             const float* __restrict__ bhhb,
                             float* __restrict__ XP) {
  extern __shared__ float a_s[];                // [64][1024] = 256 KB
  const int tid  = threadIdx.x;
  const int wv   = tid >> 5;                    // wave32
  const int lane = tid & 31;
  const int l15  = lane & 15;
  const int kk   = (lane >> 4) * 2;             // ISA A/B K sub-offset
  const int rowadd = (lane >> 4) * 8;           // C/D: lanes 16-31 hold M=r+8
  const int mb     = blockIdx.x * 64;
  const int nb_blk = blockIdx.y * 256;

  // stage A: 64 rows x 1024 cols, cooperative b128 copy
  {
    const float4* xsrc = (const float4*)(x + (size_t)mb * I_);
    float4* a4 = (float4*)a_s;
    for (int i = tid; i < 64 * I_ / 4; i += 256) a4[i] = xsrc[i];
  }
  __syncthreads();

  const int mg  = wv >> 2;                      // 0/1 -> +32 rows
  const int ng  = wv & 3;                       // 0..3 -> +64 cols
  const int nb0 = nb_blk + ng * 64;

  const float* wp[4];
#pragma unroll
  for (int nt = 0; nt < 4; ++nt) {
    int nb = nb0 + nt * 16;
    wp[nt] = ((nb < H_) ? (wf + (size_t)nb * I_)
                        : (wb + (size_t)(nb - H_) * I_)) + (size_t)l15 * I_;
  }
  const float* arow0 = a_s + (mg * 32 + l15) * I_;
  const float* arow1 = arow0 + 16 * I_;

  v8f acc[2][4] = {};
  for (int k0 = 0; k0 < I_; k0 += 4) {
    v2f a0 = *(const v2f*)(arow0 + k0 + kk);
    v2f a1 = *(const v2f*)(arow1 + k0 + kk);
#pragma unroll
    for (int nt = 0; nt < 4; ++nt) {
      v2f b = *(const v2f*)(wp[nt] + k0 + kk);
      acc[0][nt] = wmma_f32(a0, b, acc[0][nt]);
      acc[1][nt] = wmma_f32(a1, b, acc[1][nt]);
    }
  }

#pragma unroll
  for (int nt = 0; nt < 4; ++nt) {
    int col = nb0 + nt * 16 + l15;
    float bias = (col < H_) ? (bihf[col] + bhhf[col])
                            : (bihb[col - H_] + bhhb[col - H_]);
#pragma unroll
    for (int mgi = 0; mgi < 2; ++mgi)
#pragma unroll
      for (int r = 0; r < 8; ++r) {
        int row = mb + mg * 32 + mgi * 16 + r + rowadd;
        XP[(size_t)row * H2_ + col] = acc[mgi][nt][r] + bias;
      }
  }
}

// ---------------------------------------------------------------------------
// Phase 2: persistent scan. 8 blocks: dir = bid>>2, wg = bid&3 (128 N each).
// LDS: W [128][512] (256KB) + h [16][512] (32KB, rows 8-15 zero) + red (16KB).
// Waves: nhalf = wv>>2 (64-col half), ksplit = wv&3 (128-wide K slice).
// ---------------------------------------------------------------------------
__global__ void scan_kernel(const float* __restrict__ XP,
                            const float* __restrict__ whf,
                            const float* __restrict__ whb,
                            const int*   __restrict__ seqlens,
                            float* __restrict__ HB,       // [dir][buf][8][512]
                            unsigned* __restrict__ ctrs,
                            float* __restrict__ out) {    // [8][1024][1024]
  extern __shared__ float smem[];
  float* w_s = smem;                   // 128*512
  float* h_s = smem + 128 * H_;        // 16*512
  float* red = h_s + 16 * H_;          // 8 waves * 4 nt * 8 r * 16 = 4096

  const int tid  = threadIdx.x;
  const int dir  = blockIdx.x >> 2;
  const int wg   = blockIdx.x & 3;
  const int wv   = tid >> 5;
  const int lane = tid & 31;
  const int l15  = lane & 15;
  const int kk   = (lane >> 4) * 2;

  // preload W_hh slice rows [wg*128, +128) -> LDS (TDM when available)
  const float* wsrc = (dir == 0 ? whf : whb) + (size_t)(wg * 128) * H_;
#if defined(HAVE_TDM)
  if (tid < 32) {                       // wave 0 issues the tensor DMA
    tdm_load_2d_f32(wsrc, w_s, 128u, (unsigned)H_);
#if __has_builtin(__builtin_amdgcn_s_wait_tensorcnt)
    __builtin_amdgcn_s_wait_tensorcnt(0);
#endif
  }
#else
  for (int i = tid; i < 128 * H_ / 4; i += 256)
    ((float4*)w_s)[i] = ((const float4*)wsrc)[i];
#endif
  // zero h padding rows 8..15 (WMMA M-padding, stays zero)
  for (int i = tid; i < 8 * H_; i += 256) h_s[8 * H_ + i] = 0.f;

  volatile unsigned* vctr = ctrs + dir * 32;
  unsigned* actr = (unsigned*)(ctrs + dir * 32);

  for (int step = 0; step < T_; ++step) {
    const int cur = step & 1, nxt = cur ^ 1;
    const float* hbuf_cur = HB + (size_t)(dir * 2 + cur) * B_ * H_;
    float*       hbuf_nxt = HB + (size_t)(dir * 2 + nxt) * B_ * H_;

    for (int i = tid; i < B_ * H_ / 4; i += 256)
      ((float4*)h_s)[i] = ((const float4*)hbuf_cur)[i];
    __syncthreads();

    // wave (nh, ks): cols [nh*64, +64), K slice [ks*128, +128)
    const int nh = wv >> 2;
    const int kb = (wv & 3) * 128;
    v8f acc[4] = {};
    for (int k0 = kb; k0 < kb + 128; k0 += 4) {
      v2f a = *(const v2f*)(h_s + l15 * H_ + k0 + kk);
#pragma unroll
      for (int nt = 0; nt < 4; ++nt) {
        v2f b = *(const v2f*)(w_s + ((nh * 4 + nt) * 16 + l15) * H_ + k0 + kk);
        acc[nt] = wmma_f32(a, b, acc[nt]);
      }
    }
    if (lane < 16) {
#pragma unroll
      for (int nt = 0; nt < 4; ++nt)
#pragma unroll
        for (int r = 0; r < 8; ++r)
          red[((wv * 4 + nt) * 8 + r) * 16 + lane] = acc[nt][r];
    }
    __syncthreads();

    // reduce 4 K-partials, add xp, tanh, masked update + pooled store
    for (int o = tid; o < 1024; o += 256) {
      const int r = o >> 7, j = o & 127;         // batch, local n
      const int nh2 = j >> 6, nt = (j >> 4) & 3, l = j & 15;
      float s = 0.f;
#pragma unroll
      for (int ks = 0; ks < 4; ++ks) {
        const int w2 = nh2 * 4 + ks;
        s += red[((w2 * 4 + nt) * 8 + r) * 16 + l];
      }
      const int n   = wg * 128 + j;
      const int len = seqlens[r];
      const bool valid = step < len;
      float hv = h_s[r * H_ + n];                 // frozen state default
      if (valid) {
        const int tok = (dir == 0) ? step : (len - 1 - step);
        float pre = s + XP[((size_t)(r * T_ + tok)) * H2_ + dir * H_ + n];
        float hn  = fast_tanh(pre);
        hv = hn;
        if (dir == 0) {
          if (step & 1)                            // word end = 2w+1 = step
            out[((size_t)(r * NW_ + (step >> 1))) * H2_ + n] = hn;
        } else {
          if (!(tok & 1) && (tok + 1) < len)       // word start = 2w = tok
            out[((size_t)(r * NW_ + (tok >> 1))) * H2_ + H_ + n] = hn;
        }
      }
      hbuf_nxt[r * H_ + n] = hv;
    }

    __syncthreads();
    if (tid == 0) {
      __threadfence();
      atomicAdd(actr, 1u);
      const unsigned tgt = 4u * (unsigned)(step + 1);
      while (*vctr < tgt) __builtin_amdgcn_s_sleep(2);
    }
    __syncthreads();
    __threadfence();
  }
}

// ---------------------------------------------------------------------------
extern "C" void kernel_launch(void* const* d_in, const int* in_sizes, int n_in,
                              void* d_out, int out_size, void* d_ws, size_t ws_size,
                              hipStream_t stream) {
  const float* x     = (const float*)d_in[0];
  const float* w_ihf = (const float*)d_in[1];
  const float* w_hhf = (const float*)d_in[2];
  const float* b_ihf = (const float*)d_in[3];
  const float* b_hhf = (const float*)d_in[4];
  const float* w_ihb = (const float*)d_in[5];
  const float* w_hhb = (const float*)d_in[6];
  const float* b_ihb = (const float*)d_in[7];
  const float* b_hhb = (const float*)d_in[8];
  const int*   lens  = (const int*)d_in[9];

  float* ws  = (float*)d_ws;
  float* XP  = ws;                       // [8][2048][1024]
  float* HB  = ws + XP_FLOATS;           // [2][2][8][512]
  unsigned* ctrs = (unsigned*)(ws + CTR_OFF);
  float* out = (float*)d_out;            // [8][1024][1024]

  // Phase 0: zero outputs + control state
  init_kernel<<<dim3(8192), dim3(256), 0, stream>>>((float4*)out, HB);

  // Phase 1: input projection GEMM (A tile in LDS, 256 KB)
  const size_t smem1 = 64 * I_ * sizeof(float);
  xproj_kernel<<<dim3(256, 4), dim3(256), smem1, stream>>>(
      x, w_ihf, w_ihb, b_ihf, b_hhf, b_ihb, b_hhb, XP);

  // Phase 2: persistent bidirectional scan (304 KB LDS per WG)
  const size_t smem2 = (128 * H_ + 16 * H_ + 4096) * sizeof(float);
  scan_kernel<<<dim3(8), dim3(256), smem2, stream>>>(
      XP, w_hhf, w_hhb, lens, HB, ctrs, out);
}